// RetroProbe_33552284516472
// MI455X (gfx1250) — compile-verified
//
#include <hip/hip_runtime.h>
#include <hip/hip_bf16.h>
#include <math.h>

typedef __attribute__((ext_vector_type(2))) float v2f;
typedef __attribute__((ext_vector_type(8))) float v8f;

#define B_ 4
#define S_ 128
#define E_ 768
#define D_ 128
#define L_ 40

// ---------------------------------------------------------------------------
// K0: C[l] = sum_k W_lbl[l, E + k]   (tail rows use second = ones)
// ---------------------------------------------------------------------------
__global__ void wlbl_colsum_kernel(const float* __restrict__ W_lbl,
                                   float* __restrict__ Cl) {
  int l = threadIdx.x;
  if (l < L_) {
    const float* row = W_lbl + (size_t)l * (2 * E_) + E_;
    float s = 0.f;
    for (int k = 0; k < E_; ++k) s += row[k];
    Cl[l] = s;
  }
}

// ---------------------------------------------------------------------------
// K1: dep[b,s,d] = emb0[b,s,:] . W_arc[d,:]  via V_WMMA_F32_16X16X4_F32
// 256 tiles (b, tm, tn); one wave per 16x16 tile, K-loop of 192 WMMAs.
// f32 A 16x4 frag: lane<16 -> row=lane, K={0,1}; lane>=16 -> row=lane-16, K={2,3}
// f32 B 4x16 frag: lane<16 -> col=lane, K={0,1}; lane>=16 -> col=lane-16, K={2,3}
// C/D 16x16: VGPR v, M = v + 8*(lane>>4), N = lane&15
// ---------------------------------------------------------------------------
__global__ __launch_bounds__(256) void dep_gemm_kernel(
    const float* __restrict__ emb0, const float* __restrict__ W_arc,
    float* __restrict__ dep) {
  const int wave = threadIdx.x >> 5;
  const int lane = threadIdx.x & 31;
  const int tile = blockIdx.x * 8 + wave;       // 0..255
  const int b  = tile >> 6;
  const int tm = (tile >> 3) & 7;
  const int tn = tile & 7;
  const int half = lane >> 4;
  const int r    = lane & 15;

  const float* aptr = emb0 + ((size_t)b * S_ + tm * 16 + r) * E_ + half * 2;
  const float* bptr = W_arc + (size_t)(tn * 16 + r) * E_ + half * 2;

  v8f acc = {};
  for (int k = 0; k < E_; k += 4) {
    v2f a  = *(const v2f*)(aptr + k);
    v2f bb = *(const v2f*)(bptr + k);
    acc = __builtin_amdgcn_wmma_f32_16x16x4_f32(false, a, false, bb,
                                                (short)0, acc, false, false);
  }
  float* out = dep + ((size_t)b * S_ + tm * 16 + half * 8) * D_ + tn * 16 + r;
#pragma unroll
  for (int v = 0; v < 8; ++v) out[(size_t)v * D_] = acc[v];
}

// ---------------------------------------------------------------------------
// K2: AB[row, n] for row=(b*S+s) in 0..511, n in 0..79:
//   n<40 :  emb1[row,:] . W_lbl[n, 0:E]
//   n>=40:  emb1[row,:] . W_lbl[n-40, E:2E]
// (512x768)@(768x80) GEMM -> 160 tiles, one wave each.
// ---------------------------------------------------------------------------
__global__ __launch_bounds__(256) void lbl_gemm_kernel(
    const float* __restrict__ emb1, const float* __restrict__ W_lbl,
    float* __restrict__ AB) {
  const int wave = threadIdx.x >> 5;
  const int lane = threadIdx.x & 31;
  const int tile = blockIdx.x * 8 + wave;       // 0..159
  const int mt = tile / 5;                      // 0..31
  const int nt = tile % 5;                      // 0..4
  const int half = lane >> 4;
  const int r    = lane & 15;

  const float* aptr = emb1 + (size_t)(mt * 16 + r) * E_ + half * 2;
  const int n = nt * 16 + r;                    // 0..79
  const float* bbase = (n < L_) ? (W_lbl + (size_t)n * (2 * E_))
                                : (W_lbl + (size_t)(n - L_) * (2 * E_) + E_);
  const float* bptr = bbase + half * 2;

  v8f acc = {};
  for (int k = 0; k < E_; k += 4) {
    v2f a  = *(const v2f*)(aptr + k);
    v2f bb = *(const v2f*)(bptr + k);
    acc = __builtin_amdgcn_wmma_f32_16x16x4_f32(false, a, false, bb,
                                                (short)0, acc, false, false);
  }
  float* out = AB + (size_t)(mt * 16 + half * 8) * 80 + nt * 16 + r;
#pragma unroll
  for (int v = 0; v < 8; ++v) out[(size_t)v * 80] = acc[v];
}

// ---------------------------------------------------------------------------
// K3: distances[b,i,j] = ||dep[b,i]-dep[b,j]||^2 ; parent = log_softmax(-dist)
// One 128-thread block per (b,i) row; thread j owns column j.
// ---------------------------------------------------------------------------
__global__ __launch_bounds__(128) void dist_parent_kernel(
    const float* __restrict__ dep, float* __restrict__ distances,
    float* __restrict__ parent) {
  __shared__ float rowi[D_];
  __shared__ float red[128];
  const int bi = blockIdx.x;                    // b*128 + i
  const int j = threadIdx.x;
  const float* depb = dep + (size_t)(bi >> 7) * S_ * D_;
  const int i = bi & 127;

  rowi[j] = depb[(size_t)i * D_ + j];
  __syncthreads();

  const float4* rj = (const float4*)(depb + (size_t)j * D_);
  float d = 0.f;
#pragma unroll
  for (int k = 0; k < D_ / 4; ++k) {
    float4 v = rj[k];
    const float4 w = *(const float4*)(&rowi[4 * k]);
    float d0 = w.x - v.x, d1 = w.y - v.y, d2 = w.z - v.z, d3 = w.w - v.w;
    d += d0 * d0 + d1 * d1 + d2 * d2 + d3 * d3;
  }
  distances[(size_t)bi * S_ + j] = d;

  const float x = -d;
  red[j] = x;
  __syncthreads();
  for (int s = 64; s > 0; s >>= 1) {
    if (j < s) red[j] = fmaxf(red[j], red[j + s]);
    __syncthreads();
  }
  const float m = red[0];
  __syncthreads();
  red[j] = expf(x - m);
  __syncthreads();
  for (int s = 64; s > 0; s >>= 1) {
    if (j < s) red[j] += red[j + s];
    __syncthreads();
  }
  const float lse = m + logf(red[0]);
  parent[(size_t)bi * S_ + j] = x - lse;
}

// ---------------------------------------------------------------------------
// K4: epilogue. One thread per (b,r,c); writes 40 contiguous floats.
//   p = r*128+c; tail <=> r==127 (p >= S*(S-1)).
//   non-tail: i = p/127, jr = p%127, j = jr + (jr>=i)
//   mlp[l] = A[b,i,l] + (tail ? C[l] : Bm[b,j,l])
//   masked or mlp<0 -> -10 ; mlp==0 -> -FLT_MAX ; else log(mlp)+parent[b,r,c]
// ---------------------------------------------------------------------------
__global__ __launch_bounds__(256) void assemble_kernel(
    const float* __restrict__ AB, const float* __restrict__ parent,
    const float* __restrict__ Cl, const unsigned char* __restrict__ att,
    float* __restrict__ out) {
  const int t = blockIdx.x * blockDim.x + threadIdx.x;   // 0..65535
  const int b = t >> 14;
  const int r = (t >> 7) & 127;
  const int c = t & 127;
  const int p = r * S_ + c;
  const bool mask = (att[b * S_ + r] != 0) && (att[b * S_ + c] != 0) && (r != c);
  const float par = parent[(size_t)t];
  const bool tail = (r == S_ - 1);

  int i_idx, j_idx = 0;
  if (tail) {
    i_idx = c;
  } else {
    i_idx = p / 127;
    const int jr = p % 127;
    j_idx = jr + ((jr >= i_idx) ? 1 : 0);
  }
  const float* Ai = AB + ((size_t)b * S_ + i_idx) * 80;        // cols 0..39
  const float* Bj = AB + ((size_t)b * S_ + j_idx) * 80 + L_;   // cols 40..79
  float* o = out + (size_t)t * L_;

#pragma unroll 8
  for (int l = 0; l < L_; ++l) {
    const float mlp = Ai[l] + (tail ? Cl[l] : Bj[l]);
    float val;
    if (!mask || mlp < 0.f)      val = -10.0f;
    else if (mlp == 0.f)         val = -3.4028234663852886e38f;  // -FLT_MAX
    else                         val = logf(mlp) + par;
    o[l] = val;
  }
}

// ---------------------------------------------------------------------------
extern "C" void kernel_launch(void* const* d_in, const int* in_sizes, int n_in,
                              void* d_out, int out_size, void* d_ws,
                              size_t ws_size, hipStream_t stream) {
  const float* emb0 = (const float*)d_in[0];
  const float* emb1 = (const float*)d_in[1];
  const unsigned char* att = (const unsigned char*)d_in[2];  // bool (1 byte)
  const float* W_arc = (const float*)d_in[3];
  const float* W_lbl = (const float*)d_in[4];

  float* out = (float*)d_out;
  float* dep       = out;                         // 4*128*128      = 65536
  float* distances = out + 65536;                 // 4*128*128      = 65536
  float* lbl       = out + 131072;                // 4*128*128*40   = 2621440

  float* ws = (float*)d_ws;
  float* Cl     = ws;                             // 40  (padded 64)
  float* AB     = ws + 64;                        // 512*80 = 40960
  float* parent = ws + 64 + 40960;                // 4*128*128 = 65536

  wlbl_colsum_kernel<<<1, 64, 0, stream>>>(W_lbl, Cl);
  dep_gemm_kernel<<<32, 256, 0, stream>>>(emb0, W_arc, dep);
  lbl_gemm_kernel<<<20, 256, 0, stream>>>(emb1, W_lbl, AB);
  dist_parent_kernel<<<512, 128, 0, stream>>>(dep, distances, parent);
  assemble_kernel<<<256, 256, 0, stream>>>(AB, parent, Cl, att, lbl);
}